// Sorting_84894323573304
// MI455X (gfx1250) — compile-verified
//
#include <hip/hip_runtime.h>

typedef __attribute__((ext_vector_type(2))) float v2f;
typedef __attribute__((ext_vector_type(8))) float v8f;

#define BATCHES 32
#define NPB     131072            // keys per batch
#define CH      8
#define RADIX   256
#define TILE    4096              // elements per scatter/hist block
#define TILES   (NPB / TILE)      // 32 tiles per batch
#define BLK     256
#define ITEMS   16                // elements per thread in hist/scatter
#define WAVES   (BLK / 32)
#define NTOTAL  ((long long)BATCHES * (long long)NPB)   // 4,194,304 rows

// float -> sortable unsigned (ascending uint order == ascending float order)
__device__ __forceinline__ unsigned key_xform(float f) {
    unsigned u = __float_as_uint(f);
    return (u & 0x80000000u) ? ~u : (u | 0x80000000u);
}

// ---------------------------------------------------------------------------
// Kernel 1: keys[b*N+n] = xform( sum_c in[b,n,c]*w[c] ) via V_WMMA_F32_16X16X4_F32
// Each wave computes 16 rows: A = 16x4 f32 (two chained K=4 WMMAs cover C=8),
// B = w broadcast into all 16 columns -> every column of D holds the row keys.
// A layout (32-bit 16x4): lanes 0-15 = {K0,K1} in v[0],v[1]; lanes 16-31 = {K2,K3}.
// D layout (32-bit 16x16): lanes 0-15 VGPR j -> M=j ; lanes 16-31 VGPR j -> M=8+j.
// ---------------------------------------------------------------------------
__global__ void k_keys(const float* __restrict__ in, const float* __restrict__ w,
                       unsigned* __restrict__ keys, unsigned* __restrict__ idx)
{
    int wave = blockIdx.x * WAVES + (threadIdx.x >> 5);
    int lane = threadIdx.x & 31;
    int half = lane >> 4;              // 0: K={0,1}/{4,5}   1: K={2,3}/{6,7}
    int m    = lane & 15;              // row within 16-row group
    long long r0  = (long long)wave * 16;
    long long row = r0 + m;

    const float* p = in + row * CH + half * 2;
    v2f a0 = { p[0], p[1] };           // channels {0,1} or {2,3}
    v2f a1 = { p[4], p[5] };           // channels {4,5} or {6,7}
    v2f b0 = { w[half * 2 + 0], w[half * 2 + 1] };   // same for every column N
    v2f b1 = { w[half * 2 + 4], w[half * 2 + 5] };

    v8f c = {};
    c = __builtin_amdgcn_wmma_f32_16x16x4_f32(false, a0, false, b0, (short)0, c, false, false);
    c = __builtin_amdgcn_wmma_f32_16x16x4_f32(false, a1, false, b1, (short)0, c, false, false);

    // lane 0 holds M=0..7 (column 0), lane 16 holds M=8..15 (column 0)
    if (m == 0) {
        long long base = r0 + (half ? 8 : 0);
        #pragma unroll
        for (int j = 0; j < 8; ++j) {
            keys[base + j] = key_xform(c[j]);
            idx [base + j] = (unsigned)(base + j);   // global row id (batch-major)
        }
    }
}

// ---------------------------------------------------------------------------
// Kernel 2: per-(batch,tile) digit histogram -> counts[b][digit][tile]
// ---------------------------------------------------------------------------
__global__ void k_hist(const unsigned* __restrict__ keys, unsigned* __restrict__ counts,
                       int shift)
{
    __shared__ unsigned hist[RADIX];
    int t = threadIdx.x, tile = blockIdx.x, b = blockIdx.y;
    hist[t] = 0;
    __syncthreads();
    size_t base = (size_t)b * NPB + (size_t)tile * TILE;
    #pragma unroll
    for (int e = 0; e < ITEMS; ++e) {
        unsigned k = keys[base + (size_t)e * BLK + t];
        atomicAdd(&hist[(k >> shift) & 255u], 1u);
    }
    __syncthreads();
    counts[((size_t)b * RADIX + t) * TILES + tile] = hist[t];
}

// ---------------------------------------------------------------------------
// Kernel 3: per-batch exclusive scan over flat [digit][tile] (8192 entries)
// gives offs[b][d][tile] = (all smaller digits) + (same digit, earlier tiles)
// ---------------------------------------------------------------------------
__global__ void k_scan(const unsigned* __restrict__ counts, unsigned* __restrict__ offs)
{
    __shared__ unsigned s[BLK];
    int t = threadIdx.x, b = blockIdx.x;
    size_t base = (size_t)b * (RADIX * TILES);
    unsigned run = 0;
    for (int c = 0; c < (RADIX * TILES) / BLK; ++c) {
        unsigned v = counts[base + (size_t)c * BLK + t];
        __syncthreads();
        s[t] = v;
        __syncthreads();
        for (int o = 1; o < BLK; o <<= 1) {
            unsigned y = (t >= o) ? s[t - o] : 0u;
            __syncthreads();
            s[t] += y;
            __syncthreads();
        }
        unsigned incl = s[t];
        offs[base + (size_t)c * BLK + t] = incl - v + run;
        run += s[BLK - 1];
    }
}

// ---------------------------------------------------------------------------
// Kernel 4: stable scatter. Wave w owns 512 contiguous elements of the tile,
// processed as 16 wave32-synchronous rounds. Per-element rank is computed
// deterministically: match-any via 8 ballots, prefix popcount within the
// match group, leader lane bumps the per-wave per-digit LDS counter.
// ---------------------------------------------------------------------------
__global__ void k_scatter(const unsigned* __restrict__ skeys, const unsigned* __restrict__ sidx,
                          unsigned* __restrict__ dkeys, unsigned* __restrict__ didx,
                          const unsigned* __restrict__ offs, int shift)
{
    __shared__ unsigned warpHist[WAVES * RADIX];
    __shared__ unsigned offsTile[RADIX];
    int t = threadIdx.x;
    int lane = t & 31, w = t >> 5;
    int tile = blockIdx.x, b = blockIdx.y;

    for (int i = t; i < WAVES * RADIX; i += BLK) warpHist[i] = 0;
    offsTile[t] = offs[((size_t)b * RADIX + t) * TILES + tile];
    __syncthreads();

    size_t tbase = (size_t)b * NPB + (size_t)tile * TILE;
    size_t wbase = tbase + (size_t)w * (32 * ITEMS);
    unsigned ltmask = (1u << lane) - 1u;

    unsigned       kreg[ITEMS];
    unsigned short rk[ITEMS];

    #pragma unroll
    for (int e = 0; e < ITEMS; ++e) {
        unsigned k = skeys[wbase + (size_t)e * 32 + lane];
        kreg[e] = k;
        unsigned d = (k >> shift) & 255u;
        unsigned m = 0xFFFFFFFFu;
        #pragma unroll
        for (int bit = 0; bit < 8; ++bit) {
            unsigned bal = (unsigned)__builtin_amdgcn_ballot_w32(((d >> bit) & 1u) != 0u);
            m &= ((d >> bit) & 1u) ? bal : ~bal;
        }
        int leader = __ffs(m) - 1;
        unsigned prefix = __popc(m & ltmask);
        unsigned total  = __popc(m);
        unsigned old = 0;
        if (lane == leader) old = atomicAdd(&warpHist[w * RADIX + d], total);
        old = __shfl(old, leader, 32);
        rk[e] = (unsigned short)(old + prefix);
    }
    __syncthreads();

    // exclusive scan of warpHist across waves, per digit (thread t = digit t)
    {
        unsigned run = 0;
        #pragma unroll
        for (int ww = 0; ww < WAVES; ++ww) {
            unsigned c = warpHist[ww * RADIX + t];
            warpHist[ww * RADIX + t] = run;
            run += c;
        }
    }
    __syncthreads();

    size_t obase = (size_t)b * NPB;
    #pragma unroll
    for (int e = 0; e < ITEMS; ++e) {
        unsigned k = kreg[e];
        unsigned d = (k >> shift) & 255u;
        unsigned dst = offsTile[d] + warpHist[w * RADIX + d] + rk[e];
        size_t gi = wbase + (size_t)e * 32 + lane;
        dkeys[obase + dst] = k;
        didx [obase + dst] = sidx[gi];
    }
}

// ---------------------------------------------------------------------------
// Kernel 5: gather rows in sorted order (idx stores global row id)
// ---------------------------------------------------------------------------
__global__ void k_gather(const float* __restrict__ in, const unsigned* __restrict__ idx,
                         float* __restrict__ out)
{
    size_t r = (size_t)blockIdx.x * BLK + threadIdx.x;
    unsigned gi = idx[r];
    const float4* src = (const float4*)(in + (size_t)gi * CH);
    float4 a = src[0];
    float4 c = src[1];
    float4* dst = (float4*)(out + r * CH);
    dst[0] = a;
    dst[1] = c;
}

// ---------------------------------------------------------------------------
extern "C" void kernel_launch(void* const* d_in, const int* in_sizes, int n_in,
                              void* d_out, int out_size, void* d_ws, size_t ws_size,
                              hipStream_t stream)
{
    const float* in = (const float*)d_in[0];
    const float* w  = (const float*)d_in[1];
    float* out = (float*)d_out;

    char* ws = (char*)d_ws;
    const size_t NT = (size_t)NTOTAL;
    unsigned* keysA  = (unsigned*)(ws);
    unsigned* idxA   = (unsigned*)(ws + NT * 4);
    unsigned* keysB  = (unsigned*)(ws + NT * 8);
    unsigned* idxB   = (unsigned*)(ws + NT * 12);
    unsigned* counts = (unsigned*)(ws + NT * 16);
    unsigned* offs   = (unsigned*)(ws + NT * 16 + (size_t)BATCHES * RADIX * TILES * 4);

    // 1) keys via WMMA: 16 rows per wave, 8 waves per block -> 128 rows/block
    int keyBlocks = (int)(NT / (WAVES * 16));   // 32768
    k_keys<<<keyBlocks, BLK, 0, stream>>>(in, w, keysA, idxA);

    // 2) 4 LSD radix-256 passes, ping-pong A <-> B (ends back in A)
    unsigned *sk = keysA, *si = idxA, *dk = keysB, *di = idxB;
    for (int p = 0; p < 4; ++p) {
        int shift = p * 8;
        k_hist   <<<dim3(TILES, BATCHES), BLK, 0, stream>>>(sk, counts, shift);
        k_scan   <<<BATCHES,              BLK, 0, stream>>>(counts, offs);
        k_scatter<<<dim3(TILES, BATCHES), BLK, 0, stream>>>(sk, si, dk, di, offs, shift);
        unsigned* tk = sk; sk = dk; dk = tk;
        unsigned* ti = si; si = di; di = ti;
    }

    // 3) gather rows by sorted index
    k_gather<<<(int)(NT / BLK), BLK, 0, stream>>>(in, si, out);
}